// MultivariateMuyGPs_layer_77232101916859
// MI455X (gfx1250) — compile-verified
//
#include <hip/hip_runtime.h>
#include <hip/hip_bf16.h>

#define NPTS    100000
#define DIM     40
#define DIM_PAD 44         // row stride: 176 B, keeps B128 LDS writes 16-aligned,
                           // and m*44 mod 64 is distinct for m=0..15 (bank-clean)
#define BATCH   4096
#define NN      50
#define RR      10
#define NP      64         // padded rows: 50 neighbors + 1 query + zero pad
#define EPSV    1e-5f
#define SQRT3F  1.7320508075688772f
#define SQRT5F  2.23606797749979f

typedef __attribute__((ext_vector_type(2))) float v2f;
typedef __attribute__((ext_vector_type(8))) float v8f;

__device__ __forceinline__ float matern_f(float d, float inv_ls, int nu_case) {
    float r = d * inv_ls;
    if (nu_case == 0) {                 // nu = 0.5
        return __expf(-r);
    } else if (nu_case == 1) {          // nu = 1.5
        float k = SQRT3F * r;
        return (1.0f + k) * __expf(-k);
    } else {                            // nu = 2.5
        float k = SQRT5F * r;
        return (1.0f + k + k * k * (1.0f / 3.0f)) * __expf(-k);
    }
}

// Issue one wave-level async DMA: global (16B chunk) -> LDS, tracked by ASYNCcnt.
__device__ __forceinline__ void async_b128_to_lds(const void* gptr, void* lptr) {
    unsigned lds_off = (unsigned)(unsigned long long)lptr;      // low 32 bits = LDS offset
    unsigned long long gaddr = (unsigned long long)gptr;
    asm volatile("global_load_async_to_lds_b128 %0, %1, off"
                 :: "v"(lds_off), "v"(gaddr) : "memory");
}

__global__ void muygps_init_out(float* __restrict__ out) {
    if (threadIdx.x < RR) out[2 * BATCH * RR + threadIdx.x] = 0.0f;
}

__global__ __launch_bounds__(256) void muygps_kernel(
    const float* __restrict__ x,            // (N, D)
    const float* __restrict__ length_scale, // (R)
    const float* __restrict__ targets,      // (B, NN, R)
    const int*   __restrict__ binds,        // (B)
    const int*   __restrict__ nninds,       // (B, NN)
    float*       __restrict__ out)          // preds (B,R) | vars (B,R) | sigma (R)
{
    __shared__ __align__(16) float Xe[NP][DIM_PAD]; // gathered rows + query row
    __shared__ __align__(16) float Ty[NN * RR];     // this block's targets slab
    __shared__ float G [NP][NP + 1];    // Gram, then reused as distance matrix
    __shared__ float Km[NN][NN + 2];    // kernel matrix / Cholesky factor
    __shared__ float kc[NN], yv[NN], zy[NN], zc[NN];
    __shared__ float diag[NP];

    const int b    = blockIdx.x;
    const int tid  = threadIdx.x;
    const int lane = tid & 31;
    const int wave = tid >> 5;

    // ---- Stage 1a: async-DMA gather of 51 rows (510 B128 chunks) into Xe ----
    // Exactly 2 async instructions per wave.
#pragma unroll
    for (int it = 0; it < 2; ++it) {
        int e = it * 256 + tid;             // chunk id: row = e/10, quad = e%10
        if (e < (NN + 1) * 10) {
            int row = e / 10, q = e % 10;
            int srow = (row < NN) ? nninds[b * NN + row] : binds[b];
            async_b128_to_lds(x + (size_t)srow * DIM + q * 4, &Xe[row][q * 4]);
        }
    }
    // ---- Stage 1b: async-DMA this block's 500 targets (125 B128 chunks) ----
    // chunk = lane*8 + wave  =>  every wave issues exactly ONE instruction,
    // issued AFTER its x-loads (async loads complete in order per wave).
    {
        int c = lane * 8 + wave;
        if (c < (NN * RR) / 4) {
            async_b128_to_lds(targets + (size_t)b * (NN * RR) + c * 4, &Ty[c * 4]);
        }
    }
    // ---- Stage 1c: zero the pad rows 51..63 while DMAs are in flight ----
    for (int e = tid; e < (NP - NN - 1) * DIM_PAD; e += 256) {
        Xe[NN + 1 + e / DIM_PAD][e % DIM_PAD] = 0.0f;
    }
    // All but the last-issued (targets) async op complete => Xe is ready.
    asm volatile("s_wait_asynccnt 0x1" ::: "memory");
    __syncthreads();

    // ---- Stage 2: Gram matrix G = Xe * Xe^T via V_WMMA_F32_16X16X4_F32 ----
    // 16 tiles of 16x16 over the 64x64 output; 8 waves -> 2 tiles each.
    for (int t = wave; t < 16; t += 8) {
        const int tm = (t >> 2) * 16;
        const int tn = (t & 3)  * 16;
        v8f acc = {};
        const int m   = tm + (lane & 15);
        const int n   = tn + (lane & 15);
        const int khi = (lane >> 4) << 1;    // K pair selector per ISA layout
        for (int k0 = 0; k0 < DIM; k0 += 4) {
            v2f a, bb;
            a.x  = Xe[m][k0 + khi];
            a.y  = Xe[m][k0 + khi + 1];
            bb.x = Xe[n][k0 + khi];          // B = Xe^T => B[k][n] = Xe[n][k]
            bb.y = Xe[n][k0 + khi + 1];
            acc = __builtin_amdgcn_wmma_f32_16x16x4_f32(
                false, a, false, bb, (short)0, acc, false, false);
        }
        // C/D layout: VGPR v holds M = v + 8*(lane/16), N = lane%16
        const int mrow = tm + ((lane >> 4) << 3);
        const int ncol = tn + (lane & 15);
#pragma unroll
        for (int v = 0; v < 8; ++v) G[mrow + v][ncol] = acc[v];
    }
    __syncthreads();

    // ---- Stage 3: Gram -> Euclidean distances (row 50 = crosswise) ----
    if (tid < NP) diag[tid] = G[tid][tid];
    __syncthreads();
    for (int e = tid; e < NP * NP; e += 256) {
        int j = e >> 6, k = e & 63;
        float d2 = diag[j] + diag[k] - 2.0f * G[j][k];
        G[j][k] = sqrtf(fmaxf(d2, 1e-12f));
    }
    // Drain the targets DMA (overlapped with stages 2-3) before first use.
    asm volatile("s_wait_asynccnt 0x0" ::: "memory");
    __syncthreads();

    // ---- Stage 4: per-response Matern kernel + Cholesky + dual solves ----
    for (int r = 0; r < RR; ++r) {
        const float inv_ls  = 1.0f / length_scale[r];
        const int   nu_case = r % 3;   // NU_VALS pattern 0.5,1.5,2.5 repeating

        for (int e = tid; e < NN * NN; e += 256) {
            int j = e / NN, k = e % NN;
            float kv = matern_f(G[j][k], inv_ls, nu_case);
            if (j == k) kv += EPSV;
            Km[j][k] = kv;
        }
        if (tid < NN) {
            float kcr = matern_f(G[NN][tid], inv_ls, nu_case);
            float yr  = Ty[tid * RR + r];
            kc[tid] = kcr;  zc[tid] = kcr;
            yv[tid] = yr;   zy[tid] = yr;
        }
        __syncthreads();

        // Right-looking Cholesky (lower) of Km, in place.
        for (int j = 0; j < NN; ++j) {
            if (tid == 0) Km[j][j] = sqrtf(fmaxf(Km[j][j], 1e-20f));
            __syncthreads();
            const float ljj = Km[j][j];
            if (tid > j && tid < NN) Km[tid][j] /= ljj;
            __syncthreads();
            const int w = NN - 1 - j;
            for (int e = tid; e < w * w; e += 256) {
                int i = j + 1 + e / w;
                int k = j + 1 + e % w;
                if (k <= i) Km[i][k] -= Km[i][j] * Km[k][j];
            }
            __syncthreads();
        }

        // Forward solve: L z = rhs  (zy for targets, zc for Kcross)
        for (int j = 0; j < NN; ++j) {
            if (tid == 0)      zy[j] /= Km[j][j];
            else if (tid == 1) zc[j] /= Km[j][j];
            __syncthreads();
            if (tid > j && tid < NN) zy[tid] -= Km[tid][j] * zy[j];
            const int i2 = tid - 64;
            if (i2 > j && i2 < NN)   zc[i2]  -= Km[i2][j]  * zc[j];
            __syncthreads();
        }
        // Backward solve: L^T w = z
        for (int j = NN - 1; j >= 0; --j) {
            if (tid == 0)      zy[j] /= Km[j][j];
            else if (tid == 1) zc[j] /= Km[j][j];
            __syncthreads();
            if (tid < j) zy[tid] -= Km[j][tid] * zy[j];
            const int i2 = tid - 64;
            if (i2 >= 0 && i2 < j) zc[i2] -= Km[j][i2] * zc[j];
            __syncthreads();
        }

        // Reductions on wave 0
        if (wave == 0) {
            float sp = 0.0f, sv = 0.0f, ss = 0.0f;
            for (int j = lane; j < NN; j += 32) {
                sp += kc[j] * zy[j];
                sv += kc[j] * zc[j];
                ss += yv[j] * zy[j];
            }
            for (int off = 16; off > 0; off >>= 1) {
                sp += __shfl_down(sp, off);
                sv += __shfl_down(sv, off);
                ss += __shfl_down(ss, off);
            }
            if (lane == 0) {
                out[(size_t)b * RR + r]                      = sp;
                out[(size_t)BATCH * RR + (size_t)b * RR + r] = 1.0f - sv;
                atomicAdd(&out[2 * BATCH * RR + r],
                          ss * (1.0f / (float)(BATCH * NN)));
            }
        }
        __syncthreads();
    }
}

extern "C" void kernel_launch(void* const* d_in, const int* in_sizes, int n_in,
                              void* d_out, int out_size, void* d_ws, size_t ws_size,
                              hipStream_t stream) {
    const float* x            = (const float*)d_in[0];
    const float* length_scale = (const float*)d_in[1];
    const float* targets      = (const float*)d_in[2];
    const int*   binds        = (const int*)d_in[3];
    const int*   nninds       = (const int*)d_in[4];
    float*       out          = (float*)d_out;

    muygps_init_out<<<1, 32, 0, stream>>>(out);
    muygps_kernel<<<BATCH, 256, 0, stream>>>(x, length_scale, targets,
                                             binds, nninds, out);
}